// Prediction_83614423318728
// MI455X (gfx1250) — compile-verified
//
#include <hip/hip_runtime.h>
#include <stdint.h>

#define TOPK      100
#define SCALE     4.0f
#define CONF_TH   0.1f
#define SIGMA     0.5f

#define HW        65536           // 256*256
#define HM_ELEMS  (3 * HW)        // 196608 heatmap logits per slice
#define CHUNK     4096            // floats per TDM chunk (16 KB)
#define NCHUNK    (HM_ELEMS / CHUNK)  // 48
#define NBIN      4096
#define CAND_CAP  1024
#define NPB       200             // detections per batch (2 slices * 100)

typedef uint32_t u32;
typedef uint64_t u64;

typedef unsigned int v4u __attribute__((ext_vector_type(4)));
typedef int          v8i __attribute__((ext_vector_type(8)));
typedef int          v4i __attribute__((ext_vector_type(4)));

// order-preserving map: float -> uint (monotone increasing)
__device__ __forceinline__ u32 f2ord(float f) {
    u32 b = __float_as_uint(f);
    return (b & 0x80000000u) ? ~b : (b | 0x80000000u);
}
__device__ __forceinline__ float ord2f(u32 u) {
    u32 b = (u & 0x80000000u) ? (u & 0x7FFFFFFFu) : ~u;
    return __uint_as_float(b);
}

// Issue one TDM DMA: nelem contiguous f32 from global -> LDS byte offset.
// D# per CDNA5 ISA ch.8: group0 {count=1, lds_addr, global_addr[56:0], type=2},
// group1 {data_size=4B, tensor_dim0=nelem, tensor_dim1=1, tile_dim0=nelem,
//         tile_dim1=1, tensor_dim0_stride=nelem}. Groups 2/3 unused (zeros).
// Toolchain uses the 6-arg builtin: (v4u, v8i, v4i, v4i, v8i, i32 cpol).
__device__ __forceinline__ void tdm_load(const float* gsrc, u32 lds_off, u32 nelem) {
    unsigned long long ga = (unsigned long long)(uintptr_t)gsrc;
    v4u g0 = { 1u,                                        // count=1, user D#
               lds_off,                                   // lds_addr [63:32]
               (u32)(ga & 0xFFFFFFFFull),                 // global_addr[31:0]
               (u32)((ga >> 32) & 0x1FFFFFFull) | (2u << 30) }; // ga[56:32] | type=2
    v8i g1 = { (int)(2u << 16),                           // data_size=4B, wg_mask=0
               (int)((nelem & 0xFFFFu) << 16),            // tensor_dim0[15:0] @bits63:48
               (int)(((nelem >> 16) & 0xFFFFu) | (1u << 16)), // td0 hi | tensor_dim1=1
               (int)(nelem << 16),                        // tensor_dim1 hi=0 | tile_dim0
               (int)1,                                    // tile_dim1=1, tile_dim2=0
               (int)nelem,                                // tensor_dim0_stride[31:0]
               0, 0 };
    v4i g2 = { 0, 0, 0, 0 };
    v4i g3 = { 0, 0, 0, 0 };
    v8i g4 = { 0, 0, 0, 0, 0, 0, 0, 0 };
    __builtin_amdgcn_tensor_load_to_lds(g0, g1, g2, g3, g4, 0);
}

// ---------------------------------------------------------------------------
// Kernel 1: per-slice exact top-100 + decode.  grid=64, block=1024.
// ---------------------------------------------------------------------------
__global__ __launch_bounds__(1024) void topk_decode_kernel(
        const float* __restrict__ x, float* __restrict__ ws) {
    __shared__ float sbuf[2 * CHUNK];       // 32 KB TDM double buffer
    __shared__ u32   hist[NBIN];            // 16 KB
    __shared__ u64   cand[CAND_CAP];        //  8 KB
    __shared__ u32   scanb[1024];           //  4 KB
    __shared__ u32   candCnt, thrBin, jGrp;

    const int  t     = threadIdx.x;
    const int  slice = blockIdx.x;                 // 0..63  (b*2+s)
    const float* hmBase = x + (size_t)slice * 7 * HW;

    const bool wave0 = (t >> 5) == 0;
    const u32 bufOff0 = (u32)(uintptr_t)&sbuf[0];  // low 32 bits = LDS byte offset

    for (int i = t; i < NBIN; i += 1024) hist[i] = 0;
    if (t == 0) { candCnt = 0; thrBin = 0; jGrp = 0; }
    __syncthreads();

    // ---- two streaming passes over the heatmap via TDM double buffering ----
    for (int pass = 0; pass < 2; ++pass) {
        const u32 B = thrBin;                      // valid for pass 1
        if (wave0) tdm_load(hmBase, bufOff0, CHUNK);
        for (int c = 0; c < NCHUNK; ++c) {
            const int cur = c & 1;
            if (wave0) {
                if (c + 1 < NCHUNK)
                    tdm_load(hmBase + (size_t)(c + 1) * CHUNK,
                             bufOff0 + (u32)((1 - cur) * CHUNK * 4), CHUNK);
                if (c + 1 < NCHUNK) __builtin_amdgcn_s_wait_tensorcnt(1);
                else                __builtin_amdgcn_s_wait_tensorcnt(0);
            }
            __syncthreads();                       // chunk c resident in LDS
            #pragma unroll
            for (int k = 0; k < 4; ++k) {
                const int p = k * 1024 + t;
                const u32 u = f2ord(sbuf[cur * CHUNK + p]);
                const u32 bin = u >> 20;
                if (pass == 0) {
                    atomicAdd(&hist[bin], 1u);
                } else if (bin >= B) {
                    u32 pos = atomicAdd(&candCnt, 1u);
                    if (pos < CAND_CAP) {
                        u32 idx = (u32)(c * CHUNK + p);
                        cand[pos] = ((u64)u << 32) | (u32)(~idx); // desc val, asc idx
                    }
                }
            }
            __syncthreads();                       // buffer free for reuse
        }

        if (pass == 0) {
            // ---- suffix-count scan (descending bins) to find rank-100 bin ----
            const int bb = NBIN - 1 - 4 * t;
            u32 part = hist[bb] + hist[bb - 1] + hist[bb - 2] + hist[bb - 3];
            scanb[t] = part;
            __syncthreads();
            for (int off = 1; off < 1024; off <<= 1) {
                u32 v = (t >= off) ? scanb[t - off] : 0u;
                __syncthreads();
                scanb[t] += v;
                __syncthreads();
            }
            if (scanb[t] >= TOPK && (t == 0 || scanb[t - 1] < TOPK)) jGrp = (u32)t;
            __syncthreads();
            if (t == 0) {
                const u32 j = jGrp;
                u32 cum = (j > 0) ? scanb[j - 1] : 0u;
                u32 b0 = NBIN - 1 - 4 * j;
                u32 B2 = b0 - 3;
                for (int q = 0; q < 4; ++q) {
                    cum += hist[b0 - q];
                    if (cum >= TOPK) { B2 = b0 - q; break; }
                }
                thrBin = B2;
            }
            __syncthreads();
        }
    }
    __syncthreads();

    // ---- pad + exact descending bitonic sort of candidates ----
    const u32 cc = candCnt < CAND_CAP ? candCnt : CAND_CAP;
    if ((u32)t >= cc) cand[t] = 0ull;
    __syncthreads();
    for (u32 k = 2; k <= CAND_CAP; k <<= 1) {
        for (u32 j = k >> 1; j > 0; j >>= 1) {
            const u32 i = (u32)t, ixj = i ^ j;
            if (ixj > i) {
                u64 a = cand[i], b = cand[ixj];
                bool desc = ((i & k) == 0);
                if (desc ? (a < b) : (a > b)) { cand[i] = b; cand[ixj] = a; }
            }
            __syncthreads();
        }
    }

    // ---- decode ranks 0..99 ----
    if (t < TOPK) {
        const u64 key = cand[t];
        const u32 u   = (u32)(key >> 32);
        const u32 idx = ~((u32)key);
        const float logit = ord2f(u);
        float score = 1.0f / (1.0f + expf(-logit));
        if (!(score > CONF_TH)) score = 0.0f;
        const u32 cls = idx >> 16;           // idx / (H*W)
        const u32 rem = idx & 0xFFFFu;
        const float ys = (float)(rem >> 8);
        const float xs = (float)(rem & 255u);
        const float offx = hmBase[3 * HW + rem];
        const float offy = hmBase[4 * HW + rem];
        const float bw   = hmBase[5 * HW + rem] * SCALE;
        const float bh   = hmBase[6 * HW + rem] * SCALE;
        const float cx = (xs + offx) * SCALE;
        const float cy = (ys + offy) * SCALE;

        const int b  = slice >> 1, s = slice & 1;
        const int g  = b * NPB + s * TOPK + t;
        float* wsB = ws;                     // 32*200*4
        float* wsS = ws + 32 * NPB * 4;      // 32*200
        float* wsC = wsS + 32 * NPB;         // 32*200
        wsB[g * 4 + 0] = cx - bw * 0.5f;
        wsB[g * 4 + 1] = cy - bh * 0.5f;
        wsB[g * 4 + 2] = cx + bw * 0.5f;
        wsB[g * 4 + 3] = cy + bh * 0.5f;
        wsS[g] = score;
        wsC[g] = (float)cls;
    }
}

// ---------------------------------------------------------------------------
// Kernel 2: sequential soft-NMS per batch.  grid=32, block=256.
// ---------------------------------------------------------------------------
__global__ __launch_bounds__(256) void soft_nms_kernel(
        const float* __restrict__ ws, float* __restrict__ out) {
    __shared__ float bx1[NPB], by1[NPB], bx2[NPB], by2[NPB], sc[NPB], cl[NPB];
    __shared__ u64 red[256];

    const int t = threadIdx.x;
    const int b = blockIdx.x;
    const float* wsB = ws;
    const float* wsS = ws + 32 * NPB * 4;
    const float* wsC = wsS + 32 * NPB;

    if (t < NPB) {
        const int g = b * NPB + t;
        bx1[t] = wsB[g * 4 + 0]; by1[t] = wsB[g * 4 + 1];
        bx2[t] = wsB[g * 4 + 2]; by2[t] = wsB[g * 4 + 3];
        sc[t]  = wsS[g];         cl[t]  = wsC[g];
    }
    __syncthreads();

    for (int i = 0; i < NPB; ++i) {
        // argmax over positions >= i, first index wins ties (scores are >= 0)
        u64 key = 0ull;
        if (t >= i && t < NPB)
            key = ((u64)(__float_as_uint(sc[t]) | 0x80000000u) << 32) | (u32)(~(u32)t);
        red[t] = key;
        __syncthreads();
        for (int s2 = 128; s2 > 0; s2 >>= 1) {
            if (t < s2 && red[t + s2] > red[t]) red[t] = red[t + s2];
            __syncthreads();
        }
        if (t == 0) {
            const int m = (int)(~((u32)red[0]));
            if (m != i) {
                float tmp;
                tmp = bx1[i]; bx1[i] = bx1[m]; bx1[m] = tmp;
                tmp = by1[i]; by1[i] = by1[m]; by1[m] = tmp;
                tmp = bx2[i]; bx2[i] = bx2[m]; bx2[m] = tmp;
                tmp = by2[i]; by2[i] = by2[m]; by2[m] = tmp;
                tmp = sc[i];  sc[i]  = sc[m];  sc[m]  = tmp;
                tmp = cl[i];  cl[i]  = cl[m];  cl[m]  = tmp;
            }
        }
        __syncthreads();
        if (t > i && t < NPB) {
            const float ax1 = bx1[i], ay1 = by1[i], ax2 = bx2[i], ay2 = by2[i];
            const float areaI = (ax2 - ax1 + 1.0f) * (ay2 - ay1 + 1.0f);
            const float area  = (bx2[t] - bx1[t] + 1.0f) * (by2[t] - by1[t] + 1.0f);
            const float xx1 = fmaxf(ax1, bx1[t]);
            const float yy1 = fmaxf(ay1, by1[t]);
            const float xx2 = fminf(ax2, bx2[t]);
            const float yy2 = fminf(ay2, by2[t]);
            const float inter = fmaxf(0.0f, xx2 - xx1 + 1.0f) * fmaxf(0.0f, yy2 - yy1 + 1.0f);
            const float iou = inter / (areaI + area - inter);
            sc[t] *= expf(-(iou * iou) / SIGMA);
        }
        __syncthreads();
    }

    // outputs concatenated flat: boxes(32,200,4) | cls(32,200) | scores(32,200) | keep(32,200)
    if (t < NPB) {
        const int g = b * NPB + t;
        float* oB = out;
        float* oC = out + 32 * NPB * 4;
        float* oS = oC + 32 * NPB;
        float* oK = oS + 32 * NPB;
        oB[g * 4 + 0] = bx1[t]; oB[g * 4 + 1] = by1[t];
        oB[g * 4 + 2] = bx2[t]; oB[g * 4 + 3] = by2[t];
        oC[g] = cl[t];
        oS[g] = sc[t];
        oK[g] = (sc[t] > CONF_TH) ? 1.0f : 0.0f;
    }
}

extern "C" void kernel_launch(void* const* d_in, const int* in_sizes, int n_in,
                              void* d_out, int out_size, void* d_ws, size_t ws_size,
                              hipStream_t stream) {
    const float* x  = (const float*)d_in[0];
    float* ws  = (float*)d_ws;     // 32*200*(4+1+1) floats = 153600 bytes scratch
    float* out = (float*)d_out;    // 32*200*(4+1+1+1) = 44800 floats
    (void)in_sizes; (void)n_in; (void)out_size; (void)ws_size;
    topk_decode_kernel<<<64, 1024, 0, stream>>>(x, ws);
    soft_nms_kernel<<<32, 256, 0, stream>>>(ws, out);
}